// ILGR_62337155334586
// MI455X (gfx1250) — compile-verified
//
#include <hip/hip_runtime.h>
#include <hip/hip_bf16.h>

typedef __attribute__((ext_vector_type(2))) float v2f;
typedef __attribute__((ext_vector_type(8))) float v8f;

#if __has_builtin(__builtin_amdgcn_global_load_async_to_lds_b128) && \
    __has_builtin(__builtin_amdgcn_s_wait_asynccnt)
#define USE_ASYNC_LDS 1
#else
#define USE_ASYNC_LDS 0
#endif

#if USE_ASYNC_LDS
// Builtin signature (from compiler diagnostic): params are pointers to
// 'int __attribute__((vector_size(16)))' in global (AS1) / LDS (AS3) space.
typedef int v4i_mem __attribute__((vector_size(16)));
typedef __attribute__((address_space(1))) v4i_mem g_v4i;
typedef __attribute__((address_space(3))) v4i_mem l_v4i;

// generic->AS1: full 64-bit value is the global address
__device__ __forceinline__ g_v4i* to_global_v4(const void* p) {
  return (g_v4i*)(unsigned long long)(uintptr_t)p;
}
// generic->AS3: low 32 bits of a generic LDS address are the LDS offset (ISA flat addressing)
__device__ __forceinline__ l_v4i* to_lds_v4(void* p) {
  return (l_v4i*)(unsigned int)(uintptr_t)p;
}
#endif

__device__ __forceinline__ float leaky02(float x) { return x > 0.0f ? x : 0.2f * x; }

// IEEE float atomic max via signed/unsigned int ordering trick (emax seeded with a finite value first)
__device__ __forceinline__ void atomic_max_f32(float* addr, float v) {
  if (v >= 0.0f) atomicMax((int*)addr, __float_as_int(v));
  else           atomicMin((unsigned int*)addr, __float_as_uint(v));
}

__global__ void k_fill(float* __restrict__ p, float v, long n) {
  long i = (long)blockIdx.x * blockDim.x + threadIdx.x;
  if (i < n) p[i] = v;
}

// xc = [x | 1.0]  : [N,63] -> [N,64]
__global__ void k_build_xc(const float* __restrict__ x, float* __restrict__ xc, int N) {
  long t = (long)blockIdx.x * blockDim.x + threadIdx.x;
  if (t >= (long)N * 64) return;
  int i = (int)(t >> 6), f = (int)(t & 63);
  xc[t] = (f < 63) ? x[(long)i * 63 + f] : 1.0f;
}

__global__ void k_deg(const int* __restrict__ dst, float* __restrict__ deg, int E) {
  int e = blockIdx.x * blockDim.x + threadIdx.x;
  if (e < E) atomicAdd(&deg[dst[e]], 1.0f);
}

// agg[dst] += feat[src]  (one thread per edge-feature; rows are L2-resident)
__global__ void k_scatter(const int* __restrict__ src, const int* __restrict__ dst,
                          const float* __restrict__ feat, float* __restrict__ agg, int E) {
  long t = (long)blockIdx.x * blockDim.x + threadIdx.x;
  if (t >= (long)E * 64) return;
  int e = (int)(t >> 6), f = (int)(t & 63);
  atomicAdd(&agg[(long)dst[e] * 64 + f], feat[(long)src[e] * 64 + f]);
}

__global__ void k_mean(float* __restrict__ agg, const float* __restrict__ deg, int N) {
  long t = (long)blockIdx.x * blockDim.x + threadIdx.x;
  if (t >= (long)N * 64) return;
  agg[t] /= fmaxf(deg[t >> 6], 1.0f);
}

// C[N,64] = A1@W1 (+ A2@W2) (+ bias) (+ relu), all K=64, fp32 WMMA 16x16x4.
// Block = 8 waves; wave (wy,wx) computes C tile [row0+wy*16 .. +15][wx*16 .. +15].
// Weights staged to LDS via gfx1250 async global->LDS DMA when available.
__global__ void __launch_bounds__(256)
k_gemm_dual(const float* __restrict__ A1, const float* __restrict__ W1,
            const float* __restrict__ A2, const float* __restrict__ W2,
            const float* __restrict__ bias, float* __restrict__ C,
            int nrows, int relu) {
  __shared__ float sW[2][64 * 64];

#if USE_ASYNC_LDS
  {
    // 4096 floats per matrix = 1024 float4 -> 4 b128 async DMAs per thread per matrix
    for (int i4 = threadIdx.x; i4 < 1024; i4 += 256) {
      __builtin_amdgcn_global_load_async_to_lds_b128(
          to_global_v4(W1 + i4 * 4), to_lds_v4(&sW[0][i4 * 4]), 0, 0);
    }
    const float* Wb = A2 ? W2 : W1;   // harmless filler when unused
    for (int i4 = threadIdx.x; i4 < 1024; i4 += 256) {
      __builtin_amdgcn_global_load_async_to_lds_b128(
          to_global_v4(Wb + i4 * 4), to_lds_v4(&sW[1][i4 * 4]), 0, 0);
    }
    __builtin_amdgcn_s_wait_asynccnt(0);
  }
#else
  for (int i = threadIdx.x; i < 64 * 64; i += 256) {
    sW[0][i] = W1[i];
    sW[1][i] = A2 ? W2[i] : 0.0f;
  }
#endif
  __syncthreads();

  const int wave  = threadIdx.x >> 5;
  const int lane  = threadIdx.x & 31;
  const int lhalf = lane >> 4;   // 0: lanes 0-15, 1: lanes 16-31
  const int l15   = lane & 15;
  const int row0  = blockIdx.x * 32 + (wave >> 2) * 16;
  const int col0  = (wave & 3) * 16;

  // A 16x4 fp32 layout: lane half selects K pair (k+0,k+1) vs (k+2,k+3); M = lane&15
  int mA = row0 + l15;
  if (mA > nrows - 1) mA = nrows - 1;   // clamp (stores are guarded); keeps EXEC all-ones

  v8f acc = {};
  if (bias) {
    float bv = bias[col0 + l15];        // C layout: N = lane&15 for every acc element
#pragma unroll
    for (int j = 0; j < 8; ++j) acc[j] = bv;
  }

  {
    const float* Ar = A1 + (long)mA * 64;
#pragma unroll
    for (int k = 0; k < 64; k += 4) {
      const int kk = k + lhalf * 2;
      v2f a; a.x = Ar[kk]; a.y = Ar[kk + 1];
      v2f b; b.x = sW[0][kk * 64 + col0 + l15];
             b.y = sW[0][(kk + 1) * 64 + col0 + l15];
      acc = __builtin_amdgcn_wmma_f32_16x16x4_f32(false, a, false, b, (short)0, acc, false, false);
    }
  }
  if (A2) {
    const float* Ar = A2 + (long)mA * 64;
#pragma unroll
    for (int k = 0; k < 64; k += 4) {
      const int kk = k + lhalf * 2;
      v2f a; a.x = Ar[kk]; a.y = Ar[kk + 1];
      v2f b; b.x = sW[1][kk * 64 + col0 + l15];
             b.y = sW[1][(kk + 1) * 64 + col0 + l15];
      acc = __builtin_amdgcn_wmma_f32_16x16x4_f32(false, a, false, b, (short)0, acc, false, false);
    }
  }

  // C layout: VGPR j holds M = row0 + j + lhalf*8, N = col0 + (lane&15)
#pragma unroll
  for (int j = 0; j < 8; ++j) {
    int m = row0 + j + lhalf * 8;
    if (m < nrows) {
      float v = acc[j];
      if (relu) v = fmaxf(v, 0.0f);
      C[(long)m * 64 + col0 + l15] = v;
    }
  }
}

// Per-node GAT scores + self-loop score (also seeds emax, so atomicMax needs no -inf init)
__global__ void k_gat_node(const float* __restrict__ xp, const float* __restrict__ a_s,
                           const float* __restrict__ a_d, float* __restrict__ svec,
                           float* __restrict__ dvec, float* __restrict__ eself,
                           float* __restrict__ emax, int N) {
  int i = blockIdx.x * blockDim.x + threadIdx.x;
  if (i >= N) return;
  const float* r = xp + (long)i * 64;
  float s = 0.0f, d = 0.0f;
#pragma unroll 8
  for (int f = 0; f < 64; ++f) { float v = r[f]; s += v * a_s[f]; d += v * a_d[f]; }
  svec[i] = s; dvec[i] = d;
  float ee = leaky02(s + d);
  eself[i] = ee;
  emax[i] = ee;   // self-loop always hits target i -> valid seed for segment max
}

__global__ void k_edge_max(const int* __restrict__ src, const int* __restrict__ dst,
                           const float* __restrict__ svec, const float* __restrict__ dvec,
                           float* __restrict__ escore, float* __restrict__ emax, int E) {
  int e = blockIdx.x * blockDim.x + threadIdx.x;
  if (e >= E) return;
  float ee = leaky02(svec[src[e]] + dvec[dst[e]]);
  escore[e] = ee;
  atomic_max_f32(&emax[dst[e]], ee);
}

__global__ void k_self_exp(float* __restrict__ eself, const float* __restrict__ emax,
                           float* __restrict__ esum, int N) {
  int i = blockIdx.x * blockDim.x + threadIdx.x;
  if (i >= N) return;
  float ex = expf(eself[i] - emax[i]);
  eself[i] = ex;
  esum[i] = ex;   // init denom with self-loop term
}

__global__ void k_edge_exp(const int* __restrict__ dst, float* __restrict__ escore,
                           const float* __restrict__ emax, float* __restrict__ esum, int E) {
  int e = blockIdx.x * blockDim.x + threadIdx.x;
  if (e >= E) return;
  int d = dst[e];
  float ex = expf(escore[e] - emax[d]);
  escore[e] = ex;                 // reuse buffer: now holds exp(e - max)
  atomicAdd(&esum[d], ex);
}

__global__ void k_self_agg(const float* __restrict__ eself, const float* __restrict__ esum,
                           const float* __restrict__ xp, float* __restrict__ hN, int N) {
  long t = (long)blockIdx.x * blockDim.x + threadIdx.x;
  if (t >= (long)N * 64) return;
  int i = (int)(t >> 6);
  hN[t] = (eself[i] / esum[i]) * xp[t];   // full write -> no zero-init needed
}

__global__ void k_edge_agg(const int* __restrict__ src, const int* __restrict__ dst,
                           const float* __restrict__ escore, const float* __restrict__ esum,
                           const float* __restrict__ xp, float* __restrict__ hN, int E) {
  long t = (long)blockIdx.x * blockDim.x + threadIdx.x;
  if (t >= (long)E * 64) return;
  int e = (int)(t >> 6), f = (int)(t & 63);
  int d = dst[e];
  float alpha = escore[e] / esum[d];
  atomicAdd(&hN[(long)d * 64 + f], alpha * xp[(long)src[e] * 64 + f]);
}

// out = relu(cat[x1, hv1, hN1+bg1]) @ W_out + b_out
__global__ void k_out(const float* __restrict__ x1, const float* __restrict__ hv1,
                      const float* __restrict__ hN1, const float* __restrict__ bg1,
                      const float* __restrict__ W_out, const float* __restrict__ b_out,
                      float* __restrict__ out, int N) {
  int i = blockIdx.x * blockDim.x + threadIdx.x;
  if (i >= N) return;
  const float* r1 = x1  + (long)i * 64;
  const float* r2 = hv1 + (long)i * 64;
  const float* r3 = hN1 + (long)i * 64;
  float acc = b_out[0];
#pragma unroll 8
  for (int f = 0; f < 64; ++f) acc += fmaxf(r1[f], 0.0f) * W_out[f];
#pragma unroll 8
  for (int f = 0; f < 64; ++f) acc += fmaxf(r2[f], 0.0f) * W_out[64 + f];
#pragma unroll 8
  for (int f = 0; f < 64; ++f) acc += fmaxf(r3[f] + bg1[f], 0.0f) * W_out[128 + f];
  out[i] = acc;
}

extern "C" void kernel_launch(void* const* d_in, const int* in_sizes, int n_in,
                              void* d_out, int out_size, void* d_ws, size_t ws_size,
                              hipStream_t stream) {
  const float* x     = (const float*)d_in[0];
  const int*   ei    = (const int*)d_in[1];
  const float* W_l0  = (const float*)d_in[2];
  const float* W_r0  = (const float*)d_in[3];
  const float* b0    = (const float*)d_in[4];
  // d_in[5..8] = GAT layer-0 params: dead code in the reference (result discarded)
  const float* W_l1  = (const float*)d_in[9];
  const float* W_r1  = (const float*)d_in[10];
  const float* b1    = (const float*)d_in[11];
  const float* Wg1   = (const float*)d_in[12];
  const float* a_s1  = (const float*)d_in[13];
  const float* a_d1  = (const float*)d_in[14];
  const float* bg1   = (const float*)d_in[15];
  const float* W_out = (const float*)d_in[16];
  const float* b_out = (const float*)d_in[17];

  const int N = in_sizes[0] / 63;
  const int E = in_sizes[1] / 2;
  const int* src  = ei;       // edge_index[0]
  const int* dstv = ei + E;   // edge_index[1]

  float* ws = (float*)d_ws;
  const size_t NH = (size_t)N * 64;
  float* xc     = ws;              // [N,64]  (reused as xp after SAGE0)
  float* agg    = xc   + NH;       // [N,64]  scatter accumulator (reused for SAGE1)
  float* x1     = agg  + NH;       // [N,64]
  float* hN1    = x1   + NH;       // [N,64]
  float* hv1    = hN1  + NH;       // [N,64]
  float* deg    = hv1  + NH;       // [N]
  float* svec   = deg  + N;        // [N]
  float* dvec   = svec + N;        // [N]
  float* emax   = dvec + N;        // [N]
  float* eself  = emax + N;        // [N]
  float* esum   = eself + N;       // [N]
  float* escore = esum + N;        // [E]
  float* xp     = xc;              // alias

  const int T = 256;
  auto nb = [](long n) { return (int)((n + 255) / 256); };
  const long NE64 = (long)N * 64, EE64 = (long)E * 64;

  // ---- layer 0: SAGE on xc -> x1 -------------------------------------------
  k_build_xc<<<nb(NE64), T, 0, stream>>>(x, xc, N);
  k_fill<<<nb(NE64), T, 0, stream>>>(agg, 0.0f, NE64);
  k_fill<<<nb(N), T, 0, stream>>>(deg, 0.0f, N);
  k_deg<<<nb(E), T, 0, stream>>>(dstv, deg, E);
  k_scatter<<<nb(EE64), T, 0, stream>>>(src, dstv, xc, agg, E);
  k_mean<<<nb(NE64), T, 0, stream>>>(agg, deg, N);
  k_gemm_dual<<<(N + 31) / 32, 256, 0, stream>>>(agg, W_l0, xc, W_r0, b0, x1, N, 1);

  // ---- layer 1: GAT on x1 -> hN1 -------------------------------------------
  k_gemm_dual<<<(N + 31) / 32, 256, 0, stream>>>(x1, Wg1, nullptr, nullptr, nullptr, xp, N, 0);
  k_gat_node<<<nb(N), T, 0, stream>>>(xp, a_s1, a_d1, svec, dvec, eself, emax, N);
  k_edge_max<<<nb(E), T, 0, stream>>>(src, dstv, svec, dvec, escore, emax, E);
  k_self_exp<<<nb(N), T, 0, stream>>>(eself, emax, esum, N);
  k_edge_exp<<<nb(E), T, 0, stream>>>(dstv, escore, emax, esum, E);
  k_self_agg<<<nb(NE64), T, 0, stream>>>(eself, esum, xp, hN1, N);
  k_edge_agg<<<nb(EE64), T, 0, stream>>>(src, dstv, escore, esum, xp, hN1, E);

  // ---- layer 1: SAGE on x1 -> hv1 ------------------------------------------
  k_fill<<<nb(NE64), T, 0, stream>>>(agg, 0.0f, NE64);
  k_scatter<<<nb(EE64), T, 0, stream>>>(src, dstv, x1, agg, E);
  k_mean<<<nb(NE64), T, 0, stream>>>(agg, deg, N);
  k_gemm_dual<<<(N + 31) / 32, 256, 0, stream>>>(agg, W_l1, x1, W_r1, b1, hv1, N, 0);

  // ---- head -----------------------------------------------------------------
  k_out<<<nb(N), T, 0, stream>>>(x1, hv1, hN1, bg1, W_out, b_out, (float*)d_out, N);
}